// TissueExpressionHeads_23536420782761
// MI455X (gfx1250) — compile-verified
//
#include <hip/hip_runtime.h>
#include <math.h>

// ---------------------------------------------------------------------------
// TissueExpressionHeads: out[b] = softplus( gelu(g[b]·W1[t] + b1[t]) · W2[t] + b2[t] ),
// t = tissue_vector[b].  Only the selected tissue is computed (32x less work
// than the reference).  Rows are counting-sorted by tissue, then each tissue
// becomes a WMMA GEMM (M=rows_t, N=512, K=512) in exact fp32 via
// v_wmma_f32_16x16x4_f32.
// ---------------------------------------------------------------------------

#define EDIM 512
#define TNUM 32

typedef __attribute__((ext_vector_type(2))) float v2f;
typedef __attribute__((ext_vector_type(8))) float v8f;

// ---------------- prep kernels: counting sort of rows by tissue ------------

__global__ void k_init_counts(int* counts) {
    counts[threadIdx.x] = 0;
}

__global__ void k_count(const int* __restrict__ tv, int* __restrict__ counts, int Bn) {
    int b = blockIdx.x * blockDim.x + threadIdx.x;
    if (b < Bn) atomicAdd(&counts[tv[b] & (TNUM - 1)], 1);
}

__global__ void k_prefix(const int* __restrict__ counts, int* __restrict__ offsets,
                         int* __restrict__ cursors) {
    if (threadIdx.x == 0) {
        int s = 0;
        for (int t = 0; t < TNUM; ++t) {
            offsets[t] = s;
            cursors[t] = s;
            s += counts[t];
        }
        offsets[TNUM] = s;
    }
}

__global__ void k_scatter(const int* __restrict__ tv, int* __restrict__ cursors,
                          int* __restrict__ perm, int Bn) {
    int b = blockIdx.x * blockDim.x + threadIdx.x;
    if (b < Bn) {
        int pos = atomicAdd(&cursors[tv[b] & (TNUM - 1)], 1);
        perm[pos] = b;
    }
}

// ---------------- main fused GEMM + GELU + dot + softplus kernel -----------
// Block: 256 threads = 8 waves.  Block handles 64 gathered rows of one tissue
// across all 512 h-columns.  Wave w owns columns [w*64, w*64+64) as 4 n-tiles,
// and all 4 m-tiles (16 rows each): 16 accumulator tiles of v8f.

__global__ __launch_bounds__(256) void k_main(
    const float* __restrict__ g,    // (B, 512)
    const float* __restrict__ W1,   // (32, 512, 512)
    const float* __restrict__ b1,   // (32, 512)
    const float* __restrict__ W2,   // (32, 512)
    const float* __restrict__ b2,   // (32,)
    const int*  __restrict__ offsets,
    const int*  __restrict__ perm,
    float* __restrict__ out,        // (B,)
    int blocksPerT)
{
    const int t    = blockIdx.x / blocksPerT;
    const int mblk = blockIdx.x % blocksPerT;
    const int base = offsets[t];
    const int cnt  = offsets[t + 1] - base;
    if (mblk * 64 >= cnt) return;   // block-uniform early exit

    __shared__ float hsum[64];
    const int tid = threadIdx.x;
    if (tid < 64) hsum[tid] = 0.0f;
    __syncthreads();

    const int wave  = tid >> 5;
    const int lane  = tid & 31;
    const int n15   = lane & 15;          // column / row-within-tile index
    const int khalf = (lane >> 4) << 1;   // 0 for lanes 0-15, 2 for lanes 16-31

    const float* w1t = W1 + (size_t)t * EDIM * EDIM;
    const float* b1t = b1 + (size_t)t * EDIM;
    const float* w2t = W2 + (size_t)t * EDIM;

    // Per-lane A row pointers (gathered rows; clamp OOB rows to a valid one,
    // results for padded rows are never stored).
    const float* rowA[4];
#pragma unroll
    for (int mt = 0; mt < 4; ++mt) {
        int r  = mblk * 64 + mt * 16 + n15;
        int rr = (r < cnt) ? r : 0;
        rowA[mt] = g + (size_t)perm[base + rr] * EDIM;
    }

    const int n0 = wave * 64;

    v8f acc[4][4];
    const v8f vzero = {0.f, 0.f, 0.f, 0.f, 0.f, 0.f, 0.f, 0.f};
#pragma unroll
    for (int mt = 0; mt < 4; ++mt)
#pragma unroll
        for (int nt = 0; nt < 4; ++nt)
            acc[mt][nt] = vzero;

    // K loop: v_wmma_f32_16x16x4_f32, 16 WMMAs per K-step of 4.
    for (int k = 0; k < EDIM; k += 4) {
        v2f a[4];
#pragma unroll
        for (int mt = 0; mt < 4; ++mt) {
            // A frag: VGPR0 = K{k+khalf}, VGPR1 = K{k+khalf+1}; contiguous -> b64 load
            a[mt] = *(const v2f*)(rowA[mt] + k + khalf);
        }
#pragma unroll
        for (int nt = 0; nt < 4; ++nt) {
            const int col = n0 + nt * 16 + n15;
            v2f b;
            b.x = w1t[(size_t)(k + khalf) * EDIM + col];
            b.y = w1t[(size_t)(k + khalf + 1) * EDIM + col];
#pragma unroll
            for (int mt = 0; mt < 4; ++mt) {
                acc[mt][nt] = __builtin_amdgcn_wmma_f32_16x16x4_f32(
                    /*neg_a=*/false, a[mt], /*neg_b=*/false, b,
                    /*c_mod=*/(short)0, acc[mt][nt],
                    /*reuse_a=*/false, /*reuse_b=*/false);
            }
        }
    }

    // Epilogue: +b1, exact GELU, multiply by W2, reduce across 16 columns per
    // lane-half (C tile: VGPR r -> row r (lanes 0-15) / row r+8 (lanes 16-31),
    // col = lane&15), accumulate per-row sums in LDS.
#pragma unroll
    for (int mt = 0; mt < 4; ++mt) {
#pragma unroll
        for (int nt = 0; nt < 4; ++nt) {
            const int col = n0 + nt * 16 + n15;
            const float bb = b1t[col];
            const float ww = w2t[col];
#pragma unroll
            for (int r = 0; r < 8; ++r) {
                float v  = acc[mt][nt][r] + bb;
                float gv = 0.5f * v * (1.0f + erff(v * 0.70710678118654752f));
                float c  = gv * ww;
                // butterfly within each 16-lane half (masks never cross bit 4)
                c += __shfl_xor(c, 1);
                c += __shfl_xor(c, 2);
                c += __shfl_xor(c, 4);
                c += __shfl_xor(c, 8);
                if (n15 == 0) {
                    int row = mt * 16 + r + ((lane >> 4) << 3);
                    atomicAdd(&hsum[row], c);   // ds_add_f32
                }
            }
        }
    }

    __syncthreads();

    if (tid < 64) {
        int r = mblk * 64 + tid;
        if (r < cnt) {
            float x = hsum[tid] + b2[t];
            float y = (x > 20.0f) ? x : log1pf(expf(x));  // softplus
            out[perm[base + r]] = y;
        }
    }
}

// ---------------------------------------------------------------------------

extern "C" void kernel_launch(void* const* d_in, const int* in_sizes, int n_in,
                              void* d_out, int out_size, void* d_ws, size_t ws_size,
                              hipStream_t stream) {
    const float* g   = (const float*)d_in[0];
    const int*   tv  = (const int*)d_in[1];
    const float* W1  = (const float*)d_in[2];
    const float* b1v = (const float*)d_in[3];
    const float* W2  = (const float*)d_in[4];
    const float* b2v = (const float*)d_in[5];
    float* out = (float*)d_out;

    const int Bn = in_sizes[0] / EDIM;   // 8192

    // workspace layout (ints): counts[T] | offsets[T+1] | cursors[T] | perm[B]
    int* ws      = (int*)d_ws;
    int* counts  = ws;
    int* offsets = ws + TNUM;
    int* cursors = ws + 2 * TNUM + 1;
    int* perm    = ws + 3 * TNUM + 1;

    k_init_counts<<<1, TNUM, 0, stream>>>(counts);
    k_count<<<(Bn + 255) / 256, 256, 0, stream>>>(tv, counts, Bn);
    k_prefix<<<1, 1, 0, stream>>>(counts, offsets, cursors);
    k_scatter<<<(Bn + 255) / 256, 256, 0, stream>>>(tv, cursors, perm, Bn);

    const int blocksPerT = (Bn + 63) / 64;
    k_main<<<TNUM * blocksPerT, 256, 0, stream>>>(g, W1, b1v, W2, b2v,
                                                  offsets, perm, out, blocksPerT);
}